// Box2Mask_87686052315597
// MI455X (gfx1250) — compile-verified
//
#include <hip/hip_runtime.h>
#include <cmath>

#define TPB 256
#define MAXN 4096

// ---------------- Kernel 1: per-point prep ----------------
// top-2 features, per-box validity, per-view projected coords
__global__ void prep_kernel(const float* __restrict__ xyz,
                            const float* __restrict__ feats,
                            const float* __restrict__ boxes,
                            const float* __restrict__ theta,
                            const float* __restrict__ phi,
                            float2* __restrict__ coords,     // [M*N]
                            unsigned char* __restrict__ validb, // [K*N]
                            float2* __restrict__ pts,        // [N] (2nd-max, max)
                            int N, int C, int K, int M)
{
    int n = blockIdx.x * blockDim.x + threadIdx.x;
    if (n >= N) return;
    float x = xyz[3*n+0], y = xyz[3*n+1], z = xyz[3*n+2];

    // top-2 of C feature values (sorted: pts.x = 2nd largest, pts.y = largest)
    float m1 = -3.4e38f, m2 = -3.4e38f;
    for (int c = 0; c < C; ++c) {
        float v = feats[(size_t)n*C + c];
        if (v > m1) { m2 = m1; m1 = v; }
        else if (v > m2) { m2 = v; }
    }
    pts[n] = make_float2(m2, m1);

    for (int k = 0; k < K; ++k) {
        const float* b = boxes + 6*k;
        bool ok = (x >= b[0]) && (y >= b[1]) && (z >= b[2]) &&
                  (x <= b[3]) && (y <= b[4]) && (z <= b[5]);
        validb[(size_t)k*N + n] = ok ? 1 : 0;
    }

    for (int m = 0; m < M; ++m) {
        float th = theta[m], ph = phi[m];
        float st = sinf(th), ct = cosf(th), sp = sinf(ph), cp = cosf(ph);
        // center = (ct*cp, st*cp, sp) * r (r=1)
        float dx = x - ct*cp, dy = y - st*cp, dz = z - sp;
        float u = -st*dx + ct*dy;                  // U = (-st, ct, 0)
        float v = ct*sp*dx + st*sp*dy + cp*dz;     // V = (ct*sp, st*sp, cp)
        coords[(size_t)m*N + n] = make_float2(u, v);
    }
}

// ---------------- Kernel 2: per-group stats + packed point records ----------
// one block per group g = k*M + m
__global__ void build_kernel(const float2* __restrict__ coords,
                             const unsigned char* __restrict__ validb,
                             const float2* __restrict__ pts,
                             float4* __restrict__ gd,   // [G*N] (x', y', p0, p1)
                             int N, int K, int M, float res)
{
    int g = blockIdx.x;
    int k = g / M, m = g % M;
    const float2* cm = coords + (size_t)m * N;
    const unsigned char* vk = validb + (size_t)k * N;

    __shared__ float s_sx[TPB], s_sy[TPB];
    __shared__ float s_mnx[TPB], s_mxx[TPB], s_mny[TPB], s_mxy[TPB];
    __shared__ int   s_cnt[TPB];
    __shared__ float params[4];

    int tid = threadIdx.x;
    float sx = 0.f, sy = 0.f;
    float mnx = 3.4e38f, mxx = -3.4e38f, mny = 3.4e38f, mxy = -3.4e38f;
    int cnt = 0;
    for (int n = tid; n < N; n += TPB) {
        if (vk[n]) {
            float2 c = cm[n];
            sx += c.x; sy += c.y;
            mnx = fminf(mnx, c.x); mxx = fmaxf(mxx, c.x);
            mny = fminf(mny, c.y); mxy = fmaxf(mxy, c.y);
            cnt++;
        }
    }
    s_sx[tid] = sx; s_sy[tid] = sy;
    s_mnx[tid] = mnx; s_mxx[tid] = mxx; s_mny[tid] = mny; s_mxy[tid] = mxy;
    s_cnt[tid] = cnt;
    __syncthreads();

    if (tid == 0) {
        float SX = 0.f, SY = 0.f;
        float MNX = 3.4e38f, MXX = -3.4e38f, MNY = 3.4e38f, MXY = -3.4e38f;
        int CNT = 0;
        for (int t = 0; t < TPB; ++t) {
            SX += s_sx[t]; SY += s_sy[t];
            MNX = fminf(MNX, s_mnx[t]); MXX = fmaxf(MXX, s_mxx[t]);
            MNY = fminf(MNY, s_mny[t]); MXY = fmaxf(MXY, s_mxy[t]);
            CNT += s_cnt[t];
        }
        float meanx = SX / (float)CNT;
        float meany = SY / (float)CNT;
        // reference: cmin/cmax over coords with invalid rows replaced by mean_c
        if (CNT < N) {
            MNX = fminf(MNX, meanx); MXX = fmaxf(MXX, meanx);
            MNY = fminf(MNY, meany); MXY = fmaxf(MXY, meany);
        }
        params[0] = (MXX + MNX) * 0.5f;                 // center x
        params[1] = fmaxf(MXX - MNX, 1e-5f) * 0.5f;     // scale  x
        params[2] = (MXY + MNY) * 0.5f;                 // center y
        params[3] = fmaxf(MXY - MNY, 1e-5f) * 0.5f;     // scale  y
    }
    __syncthreads();

    float ctr0 = params[0], scl0 = params[1], ctr1 = params[2], scl1 = params[3];
    float a = 0.4f * res;   // 0.8 * p / 2
    float b = 0.1f * res;
    for (int n = tid; n < N; n += TPB) {
        float4 o;
        if (vk[n]) {
            float2 c = cm[n];
            float2 p = pts[n];
            o.x = ((c.x - ctr0) / scl0 + 1.0f) * a + b;
            o.y = ((c.y - ctr1) / scl1 + 1.0f) * a + b;
            o.z = p.x; o.w = p.y;
        } else {
            // invalid points can never satisfy within=(d2<9)&valid -> sentinel
            o.x = 1e9f; o.y = 1e9f; o.z = 0.f; o.w = 0.f;
        }
        gd[(size_t)g * N + n] = o;
    }
}

// ---------------- Kernel 3: rasterize ----------------
// One block per (group, 256-pixel tile). Stage the group's 64KB point table
// into LDS with gfx1250 async global->LDS (ASYNCcnt), then ordered scan.
__global__ void raster_kernel(const float4* __restrict__ gd,
                              float* __restrict__ out,
                              int N, int res, int tilesPerG)
{
    __shared__ float4 smem[MAXN];
    int g    = blockIdx.x / tilesPerG;
    int tile = blockIdx.x % tilesPerG;
    const float4* src = gd + (size_t)g * N;

    // LDS byte offset of smem (generic shared ptr low 32 bits = LDS offset)
    unsigned ldsBase = (unsigned)(size_t)(&smem[0]);
    for (int e = threadIdx.x; e < N; e += TPB) {
        unsigned long long gaddr = (unsigned long long)(const void*)(src + e);
        unsigned lo = ldsBase + (unsigned)e * 16u;
        asm volatile("global_load_async_to_lds_b128 %0, %1, off"
                     :: "v"(lo), "v"(gaddr) : "memory");
    }
    asm volatile("s_wait_asynccnt 0x0" ::: "memory");
    __syncthreads();

    int s = tile * TPB + threadIdx.x;
    if (s < res * res) {
        int i = s / res, j = s % res;
        float fi = (float)i, fj = (float)j;
        float sum0 = 0.f, sum1 = 0.f;
        int c0 = 0, c1 = 0, found = 0;
        // first 16 within-radius points in ascending index order
        for (int n = 0; n < N; ++n) {
            float4 p = smem[n];
            float dx = fi - p.x, dy = fj - p.y;
            float d2 = dx*dx + dy*dy;
            if (d2 < 9.0f) {            // RADIUS^2, strict <, matches ref
                sum0 += p.z; sum1 += p.w;
                if (p.z != 0.0f) c0++;
                if (p.w != 0.0f) c1++;
                if (++found == 16) break;
            }
        }
        float m0 = sum0 / (float)(c0 ? c0 : 1);
        float m1 = sum1 / (float)(c1 ? c1 : 1);
        float mx = fmaxf(m0, m1);
        float e0 = expf(m0 - mx), e1 = expf(m1 - mx);
        float inv = 1.0f / (e0 + e1);
        float nf0 = e0 * inv, nf1 = e1 * inv;
        float val = (nf0 == nf1) ? 0.0f : nf1 * 255.0f;

        size_t plane = (size_t)res * res;
        size_t base  = ((size_t)g * 3) * plane + (size_t)i * res + j;
        out[base]           = val;
        out[base +   plane] = val;
        out[base + 2*plane] = val;
    }
}

extern "C" void kernel_launch(void* const* d_in, const int* in_sizes, int n_in,
                              void* d_out, int out_size, void* d_ws, size_t ws_size,
                              hipStream_t stream)
{
    const float* xyz   = (const float*)d_in[0];
    const float* feats = (const float*)d_in[1];
    const float* boxes = (const float*)d_in[2];
    const float* theta = (const float*)d_in[3];
    const float* phi   = (const float*)d_in[4];
    // d_in[5] = res (device scalar); recover res from out_size instead.

    int N = in_sizes[0] / 3;          // B = 1
    int C = in_sizes[1] / N;
    int K = in_sizes[2] / 6;
    int M = in_sizes[3];
    int G = K * M;
    int res = (int)(std::sqrt((double)out_size / (3.0 * (double)G)) + 0.5);

    char* ws = (char*)d_ws;
    size_t off = 0;
    float2* pts = (float2*)(ws + off);          off += (size_t)N * sizeof(float2);
    float2* coords = (float2*)(ws + off);       off += (size_t)M * N * sizeof(float2);
    unsigned char* validb = (unsigned char*)(ws + off); off += (size_t)K * N;
    off = (off + 15) & ~(size_t)15;
    float4* gd = (float4*)(ws + off);           off += (size_t)G * N * sizeof(float4);

    prep_kernel<<<(N + TPB - 1) / TPB, TPB, 0, stream>>>(
        xyz, feats, boxes, theta, phi, coords, validb, pts, N, C, K, M);

    build_kernel<<<G, TPB, 0, stream>>>(coords, validb, pts, gd, N, K, M, (float)res);

    int tilesPerG = (res * res + TPB - 1) / TPB;
    raster_kernel<<<G * tilesPerG, TPB, 0, stream>>>(gd, (float*)d_out, N, res, tilesPerG);
}